// APPNP_Net_52261162057816
// MI455X (gfx1250) — compile-verified
//
#include <hip/hip_runtime.h>
#include <math.h>

// APPNP_Net on MI455X (gfx1250).
// ALPHA == 1.0  =>  z_{t+1} = 0*agg + 1*h = h exactly (agg is finite since
// self-loops force deg>=1), so the K-step propagation is the identity and
// edge_index is dead.  out = log_softmax(relu(x@W1+b1)@W2+b2).
//
// Fully fused: one workgroup computes 64 rows end-to-end. h (100000x256,
// 102MB) never hits HBM; only x (204.8MB) + out (20MB) + L2-resident weights
// move, ~9.7us of HBM traffic at 23.3TB/s. Matrix math uses native f32 WMMA
// (V_WMMA_F32_16X16X4_F32) to preserve the f32 reference numerics.

#define N_NODES 100000
#define F_IN    512
#define HID     256
#define C_OUT   50
#define MBLK    64
#define KC      64

typedef float v2f __attribute__((ext_vector_type(2)));
typedef float v8f __attribute__((ext_vector_type(8)));

// LDS plan (bytes):
//   Xs : [64][68]  f32     = 17408   (reused as Zs[64][68] in epilogue)
//   Wp : [32][272] float2  = 69632   (GEMM1 W1 pack; reused as [32][80] for W2)
//   Hs : [64][260] f32     = 66560
//   total                   = 153600  (<320KB WGP LDS)
#define XS_OFF  0
#define WP_OFF  4352          // floats
#define HS_OFF  (4352 + 17408)
#define SMEM_FLOATS 38400     // 153600 B

__global__ __launch_bounds__(256)
void appnp_fused(const float* __restrict__ x,
                 const float* __restrict__ W1, const float* __restrict__ b1,
                 const float* __restrict__ W2, const float* __restrict__ b2,
                 float* __restrict__ out)
{
    __shared__ __align__(16) float smem[SMEM_FLOATS];
    float* Xs = smem + XS_OFF;            // [64][68]
    v2f*   Wp = (v2f*)(smem + WP_OFF);    // phase1: [32][272], phase2: [32][80]
    float* Hs = smem + HS_OFF;            // [64][260]

    const int tid  = threadIdx.x;
    const int lane = tid & 31;
    const int wv   = tid >> 5;            // 0..7
    const int half = lane >> 4;           // 0/1 : K split across half-waves
    const int l16  = lane & 15;
    const int m0   = blockIdx.x * MBLK;

    // ---------------- Phase 1: Hs = relu(X @ W1 + b1) ----------------
    // wave -> 16x128 slab: M-tile = wv%4, N-half = wv/4, 8 accumulators.
    const int mrow   = (wv & 3) * 16;
    const int nbase0 = (wv >> 2) * 128;

    v8f acc[8];
    #pragma unroll
    for (int t = 0; t < 8; ++t)
        #pragma unroll
        for (int i = 0; i < 8; ++i) acc[t][i] = 0.0f;

    for (int kc = 0; kc < F_IN; kc += KC) {
        // stage X tile: Xs[m][k], row pad 68 -> bank (4m+k)%64, conflict-free
        {
            const int k  = tid & 63;
            const int mq = tid >> 6;                       // 0..3
            #pragma unroll
            for (int m = mq; m < MBLK; m += 4) {
                int gm = m0 + m; if (gm >= N_NODES) gm = N_NODES - 1; // clamp
                Xs[m * 68 + k] = x[(long)gm * F_IN + kc + k];
            }
        }
        // stage W1 tile, K-pair packed: Wp[k2][n] = {W1[kc+2k2][n], W1[kc+2k2+1][n]}
        {
            const int n = tid;                              // 0..255
            #pragma unroll
            for (int k2 = 0; k2 < 32; ++k2) {
                v2f w;
                w.x = W1[(kc + 2 * k2)     * HID + n];
                w.y = W1[(kc + 2 * k2 + 1) * HID + n];
                Wp[k2 * 272 + n] = w;
            }
        }
        __syncthreads();

        #pragma unroll
        for (int ks = 0; ks < KC / 4; ++ks) {
            // A frag: lane(l16,half) -> A[M=mrow+l16][K = 4ks+2*half +{0,1}]
            v2f a = *(const v2f*)&Xs[(mrow + l16) * 68 + 4 * ks + 2 * half];
            const int k2 = 2 * ks + half;
            #pragma unroll
            for (int t = 0; t < 8; ++t) {
                v2f b = Wp[k2 * 272 + nbase0 + 16 * t + l16];
                acc[t] = __builtin_amdgcn_wmma_f32_16x16x4_f32(
                    false, a, false, b, (short)0, acc[t], false, false);
            }
        }
        __syncthreads();
    }

    // spill h tile to LDS with bias+relu; D layout: row = v + 8*half
    #pragma unroll
    for (int t = 0; t < 8; ++t) {
        const int n = nbase0 + 16 * t + l16;
        const float bias = b1[n];
        #pragma unroll
        for (int v = 0; v < 8; ++v) {
            const int m = mrow + v + 8 * half;
            float hv = acc[t][v] + bias;
            Hs[m * 260 + n] = hv > 0.0f ? hv : 0.0f;
        }
    }
    __syncthreads();

    // ---------------- Phase 2: Z = Hs @ W2 + b2  (N padded 50->64) ----------
    const int mrow2  = (wv & 3) * 16;
    const int nbase2 = (wv >> 2) * 32;                      // two 16-wide tiles

    v8f zacc[2];
    #pragma unroll
    for (int t = 0; t < 2; ++t)
        #pragma unroll
        for (int i = 0; i < 8; ++i) zacc[t][i] = 0.0f;

    for (int kc = 0; kc < HID; kc += KC) {
        // stage W2 pack into reused Wp space: [32][80] float2, zero pad n>=50
        {
            const int n   = tid & 63;
            const int k2q = tid >> 6;                       // 0..3
            #pragma unroll
            for (int k2 = k2q; k2 < 32; k2 += 4) {
                v2f w; w.x = 0.0f; w.y = 0.0f;
                if (n < C_OUT) {
                    w.x = W2[(kc + 2 * k2)     * C_OUT + n];
                    w.y = W2[(kc + 2 * k2 + 1) * C_OUT + n];
                }
                Wp[k2 * 80 + n] = w;
            }
        }
        __syncthreads();

        #pragma unroll
        for (int ks = 0; ks < KC / 4; ++ks) {
            v2f a = *(const v2f*)&Hs[(mrow2 + l16) * 260 + kc + 4 * ks + 2 * half];
            const int k2 = 2 * ks + half;
            #pragma unroll
            for (int t = 0; t < 2; ++t) {
                v2f b = Wp[k2 * 80 + nbase2 + 16 * t + l16];
                zacc[t] = __builtin_amdgcn_wmma_f32_16x16x4_f32(
                    false, a, false, b, (short)0, zacc[t], false, false);
            }
        }
        __syncthreads();
    }

    // z + b2 into Zs (reuse Xs region, [64][68])
    float* Zs = Xs;
    #pragma unroll
    for (int t = 0; t < 2; ++t) {
        const int n = nbase2 + 16 * t + l16;
        const float bias = (n < C_OUT) ? b2[n] : 0.0f;
        #pragma unroll
        for (int v = 0; v < 8; ++v) {
            const int m = mrow2 + v + 8 * half;
            Zs[m * 68 + n] = zacc[t][v] + bias;
        }
    }
    __syncthreads();

    // ---------------- Phase 3: row-wise log_softmax (wave per row) ----------
    for (int m = wv; m < MBLK; m += 8) {
        const int gm = m0 + m;
        const float v0 = Zs[m * 68 + lane];                 // lane < 32 < 50
        const int  n1  = lane + 32;
        const bool has1 = (n1 < C_OUT);
        const float v1 = has1 ? Zs[m * 68 + n1] : -INFINITY;

        float mx = fmaxf(v0, v1);
        #pragma unroll
        for (int off = 16; off > 0; off >>= 1)
            mx = fmaxf(mx, __shfl_xor(mx, off, 32));

        float s = expf(v0 - mx) + (has1 ? expf(v1 - mx) : 0.0f);
        #pragma unroll
        for (int off = 16; off > 0; off >>= 1)
            s += __shfl_xor(s, off, 32);

        const float lse = mx + logf(s);
        if (gm < N_NODES) {
            out[gm * C_OUT + lane] = v0 - lse;
            if (has1) out[gm * C_OUT + n1] = v1 - lse;
        }
    }
}

extern "C" void kernel_launch(void* const* d_in, const int* in_sizes, int n_in,
                              void* d_out, int out_size, void* d_ws, size_t ws_size,
                              hipStream_t stream) {
    const float* x  = (const float*)d_in[0];
    // d_in[1] = edge_index: unused — ALPHA==1.0 makes APPNP propagation identity.
    const float* W1 = (const float*)d_in[2];
    const float* b1 = (const float*)d_in[3];
    const float* W2 = (const float*)d_in[4];
    const float* b2 = (const float*)d_in[5];
    float* out = (float*)d_out;

    const int blocks = (N_NODES + MBLK - 1) / MBLK;   // 1563
    appnp_fused<<<blocks, 256, 0, stream>>>(x, W1, b1, W2, b2, out);
}